// SEBlock_88373247082503
// MI455X (gfx1250) — compile-verified
//
#include <hip/hip_runtime.h>
#include <cstdint>

// Problem constants (from reference setup_inputs)
#define GRIDN 3
#define BATCH 8
#define CCH   64          // c
#define HH    128
#define WW    128
#define H0    64
#define W0    64
#define DCH   (GRIDN * 4 * CCH)   // 768 depth channels

#define TILE  16          // output tile edge
#define SRC   10          // source window edge per tile (2x upsample + 1 halo each side)
#define NCH   48          // 3 grids * 16 (i,j) combos for one group q
#define LDS_ELEMS (NCH * SRC * SRC)   // 4800 floats = 19.2 KB

#ifndef __has_builtin
#define __has_builtin(x) 0
#endif

#if __has_builtin(__builtin_amdgcn_global_load_async_to_lds_b32)
#define ASYNC_BUILTIN 1
#endif
#if __has_builtin(__builtin_amdgcn_s_wait_asynccnt)
#define WAITASYNC_BUILTIN 1
#endif

typedef __attribute__((address_space(1))) int g1_int;
typedef __attribute__((address_space(3))) int l3_int;

__device__ __forceinline__ void async_copy_f32(const float* gptr, const float* lptr) {
#ifdef ASYNC_BUILTIN
  __builtin_amdgcn_global_load_async_to_lds_b32(
      (g1_int*)(uintptr_t)gptr,
      (l3_int*)(unsigned)(uintptr_t)lptr,
      0, 0);
#else
  // LDS flat addresses: low 32 bits are the LDS byte offset (ISA 10.2).
  unsigned lds_off = (unsigned)(uintptr_t)lptr;
  asm volatile("global_load_async_to_lds_b32 %0, %1, off"
               :: "v"(lds_off), "v"(gptr) : "memory");
#endif
}

__device__ __forceinline__ void wait_asynccnt0() {
#ifdef WAITASYNC_BUILTIN
  __builtin_amdgcn_s_wait_asynccnt(0);
#else
  asm volatile("s_wait_asynccnt 0" ::: "memory");
#endif
}

__global__ __launch_bounds__(256)
void depth_mix_fused_kernel(const float* __restrict__ xin,
                            const float* __restrict__ depth,
                            const float* __restrict__ dori,
                            float* __restrict__ out)
{
  __shared__ float tile[LDS_ELEMS];

  const int tid    = threadIdx.x;
  const int blk    = blockIdx.x;
  const int tileId = blk & 63;          // 8x8 tiles over 128x128
  const int q      = (blk >> 6) & 15;   // channel group 0..15
  const int b      = blk >> 10;         // batch 0..7

  const int tileY = (tileId >> 3) * TILE;
  const int tileX = (tileId & 7)  * TILE;
  const int sy0   = (tileY >> 1) - 1;   // source window origin (may be -1)
  const int sx0   = (tileX >> 1) - 1;

  // Per-thread output pixel (also used to prefetch x while async copies fly)
  const int py = tid >> 4, px = tid & 15;
  const int y  = tileY + py, x = tileX + px;
  const size_t pix = (size_t)y * WW + x;
  const float* xb  = xin + ((size_t)b * CCH + q) * (HH * WW) + pix;
  __builtin_prefetch(xb, 0, 0);   // global_prefetch_b8 -> warm L2 for phase 3

  // ---- Phase 1: async-stage 48 depth channels' 10x10 clamped windows into LDS
  const float* dbase = depth + (size_t)b * DCH * H0 * W0;
  for (int e = tid; e < LDS_ELEMS; e += 256) {
    int chl  = e / 100;           // 0..47  (g*16 + i*4 + j)
    int pos  = e - chl * 100;
    int r    = pos / 10;
    int cc   = pos - r * 10;
    int g    = chl >> 4;
    int ij   = chl & 15;
    int i4   = ij >> 2;
    int j4   = ij & 3;
    int ch   = g * 256 + i4 * 64 + j4 * 16 + q;
    int row  = sy0 + r;  row = row < 0 ? 0 : (row > H0 - 1 ? H0 - 1 : row);
    int col  = sx0 + cc; col = col < 0 ? 0 : (col > W0 - 1 ? W0 - 1 : col);
    const float* gp = dbase + ((size_t)ch * H0 + row) * W0 + col;
    async_copy_f32(gp, &tile[e]);
  }
  wait_asynccnt0();
  __syncthreads();

  // ---- Phase 2: sigmoid in place (once per source element, not per tap)
  for (int e = tid; e < LDS_ELEMS; e += 256) {
    float v = tile[e];
    tile[e] = __fdividef(1.0f, 1.0f + __expf(-v));
  }
  __syncthreads();

  // ---- Phase 3: bilerp + depth-weighted 4x4 mix, one pixel per thread
  // half-pixel 2x upsample: s = y*0.5 - 0.25
  const float syf = (float)y * 0.5f - 0.25f;
  const float sxf = (float)x * 0.5f - 0.25f;
  const int  iy0  = (int)floorf(syf);
  const int  ix0  = (int)floorf(sxf);
  const float fy  = syf - (float)iy0;
  const float fx  = sxf - (float)ix0;
  const int  ly0  = iy0 - sy0;          // 0..8 (clamped slots handle edges)
  const int  lx0  = ix0 - sx0;          // 0..8
  const float w00 = (1.0f - fy) * (1.0f - fx);
  const float w01 = (1.0f - fy) * fx;
  const float w10 = fy * (1.0f - fx);
  const float w11 = fy * fx;

  const float dv  = dori[((size_t)b * HH + y) * WW + x] * 3.0f;
  float rdv[GRIDN];
  #pragma unroll
  for (int g = 0; g < GRIDN; ++g)
    rdv[g] = fmaxf(1.0f - fabsf((float)g - dv), 0.0f);

  float xv[4];
  #pragma unroll
  for (int j = 0; j < 4; ++j)
    xv[j] = xb[(size_t)j * 16 * HH * WW];

  float acc[4] = {xv[0], xv[1], xv[2], xv[3]};   // identity term
  const int base00 = ly0 * 10 + lx0;
  #pragma unroll
  for (int g = 0; g < GRIDN; ++g) {
    const float rg = rdv[g];
    #pragma unroll
    for (int i = 0; i < 4; ++i) {
      #pragma unroll
      for (int j = 0; j < 4; ++j) {
        const float* t = &tile[(g * 16 + i * 4 + j) * 100 + base00];
        float v = w00 * t[0] + w01 * t[1] + w10 * t[10] + w11 * t[11];
        acc[i] = __fmaf_rn(rg * v, xv[j], acc[i]);
      }
    }
  }

  float* ob = out + ((size_t)b * CCH + q) * (HH * WW) + pix;
  #pragma unroll
  for (int i = 0; i < 4; ++i)
    ob[(size_t)i * 16 * HH * WW] = acc[i];
}

extern "C" void kernel_launch(void* const* d_in, const int* in_sizes, int n_in,
                              void* d_out, int out_size, void* d_ws, size_t ws_size,
                              hipStream_t stream) {
  (void)in_sizes; (void)n_in; (void)out_size; (void)d_ws; (void)ws_size;
  const float* x     = (const float*)d_in[0];
  const float* depth = (const float*)d_in[1];
  const float* dori  = (const float*)d_in[2];
  float* out = (float*)d_out;

  const int nBlocks = BATCH * 16 * 64;   // batch * groups * tiles = 8192
  depth_mix_fused_kernel<<<nBlocks, 256, 0, stream>>>(x, depth, dori, out);
}